// ChamferFunctionWithIdxNoGrad_78383153152278
// MI455X (gfx1250) — compile-verified
//
#include <hip/hip_runtime.h>

// CDNA5 / gfx1250 chamfer bidirectional NN using V_WMMA_F32_16X16X4_F32.
// d(q,p) = |q|^2 + |p|^2 - 2 q.p ; fold |p|^2 into the K=4 GEMM:
//   A[m,:] = [-2x, -2y, -2z, |p|^2]   (database tile, 16x4)
//   B[:,n] = [qx, qy, qz, 1]          (query tile,     4x16)
// => D[m,n] = |p_m|^2 - 2 q_n.p_m   (add |q_n|^2 once at the end)

typedef __attribute__((ext_vector_type(2))) float v2f;
typedef __attribute__((ext_vector_type(8))) float v8f;

#define QT 2  // query tiles (of 16 queries) per wave

// Scatter database points into the WMMA 32-bit A-matrix (16x4) VGPR layout:
// lane r (0..15):  V0 = A[r,0] = -2x, V1 = A[r,1] = -2y
// lane r+16:       V0 = A[r,2] = -2z, V1 = A[r,3] = |p|^2
// stored as Aprep[tile][lane][2] floats (64 floats / tile).
__global__ void chamfer_prep(const float* __restrict__ xyz,
                             float* __restrict__ Aprep, int total) {
  int i = blockIdx.x * blockDim.x + threadIdx.x;
  if (i >= total) return;
  float x = xyz[3 * i + 0];
  float y = xyz[3 * i + 1];
  float z = xyz[3 * i + 2];
  float sq = __builtin_fmaf(x, x, __builtin_fmaf(y, y, z * z));
  int tile = i >> 4;
  int r = i & 15;
  float* base = Aprep + (size_t)tile * 64;
  base[2 * r + 0] = -2.0f * x;
  base[2 * r + 1] = -2.0f * y;
  base[2 * (r + 16) + 0] = -2.0f * z;
  base[2 * (r + 16) + 1] = sq;
}

// ds_swizzle SWAPX16: xor=0x10, or=0, and=0x1f -> offset 0x401F (lane l <-> l^16)
__device__ __forceinline__ float swapx16_f(float v) {
  int i = __builtin_bit_cast(int, v);
  i = __builtin_amdgcn_ds_swizzle(i, 0x401F);
  return __builtin_bit_cast(float, i);
}
__device__ __forceinline__ int swapx16_i(int v) {
  return __builtin_amdgcn_ds_swizzle(v, 0x401F);
}

__global__ void __launch_bounds__(256)
chamfer_nn(const float* __restrict__ Aprep,  // prepped database, [B*ntiles][64]
           const float* __restrict__ qxyz,   // raw query points  [B][Nq][3]
           float* __restrict__ outd, int* __restrict__ outi,
           int Mdb, int Nq) {
  const int lane = threadIdx.x & 31;
  const int wave = blockIdx.x * (blockDim.x >> 5) + (threadIdx.x >> 5);
  const int wavesPerBatch = Nq / (16 * QT);
  const int b = wave / wavesPerBatch;
  const int qbase = (wave - b * wavesPerBatch) * (16 * QT);
  const int col = lane & 15;
  const bool hi = lane >= 16;

  // Build B operand (4x16 f32): V0 = {qx | qz}, V1 = {qy | 1.0} across halves.
  v2f breg[QT];
  float sqq[QT];
#pragma unroll
  for (int t = 0; t < QT; ++t) {
    int n = qbase + t * 16 + col;
    const float* qp = qxyz + ((size_t)b * Nq + n) * 3;
    float qx = qp[0], qy = qp[1], qz = qp[2];
    sqq[t] = __builtin_fmaf(qx, qx, __builtin_fmaf(qy, qy, qz * qz));
    breg[t].x = hi ? qz : qx;
    breg[t].y = hi ? 1.0f : qy;
  }

  // Running min per C/D slot; slot (lane,v) always has m%16 == v + 8*hi,
  // so tracking the winning *tile* index is enough to reconstruct argmin.
  v8f acc[QT];
  int bt[QT][8];
#pragma unroll
  for (int t = 0; t < QT; ++t)
#pragma unroll
    for (int v = 0; v < 8; ++v) {
      acc[t][v] = 3.0e38f;
      bt[t][v] = 0;
    }

  const int ntiles = Mdb >> 4;
  const float* aPtr = Aprep + (size_t)b * ntiles * 64 + lane * 2;
  for (int tile = 0; tile < ntiles; ++tile) {
    v2f a = *(const v2f*)aPtr;  // global_load_b64, A operand for this tile
    aPtr += 64;
#pragma unroll
    for (int t = 0; t < QT; ++t) {
      v8f c = {0.f, 0.f, 0.f, 0.f, 0.f, 0.f, 0.f, 0.f};
      // D[m,n] = |p_m|^2 - 2 q_n.p_m  (16x16 tile in one matrix op)
      c = __builtin_amdgcn_wmma_f32_16x16x4_f32(
          false, a, false, breg[t], (short)0, c, false, false);
#pragma unroll
      for (int v = 0; v < 8; ++v) {
        bool lt = c[v] < acc[t][v];
        acc[t][v] = lt ? c[v] : acc[t][v];
        bt[t][v] = lt ? tile : bt[t][v];  // tile is uniform (SGPR cndmask)
      }
    }
  }

  // Final reduction: 8 slots in-lane, then cross-half swap (lanes l <-> l+16).
  const int half = hi ? 8 : 0;
#pragma unroll
  for (int t = 0; t < QT; ++t) {
    float dbest = acc[t][0];
    int mbest = bt[t][0] * 16 + half;
#pragma unroll
    for (int v = 1; v < 8; ++v) {
      float d = acc[t][v];
      int m = bt[t][v] * 16 + v + half;
      bool take = (d < dbest) || (d == dbest && m < mbest);
      dbest = take ? d : dbest;
      mbest = take ? m : mbest;
    }
    float d2 = swapx16_f(dbest);
    int m2 = swapx16_i(mbest);
    bool take = (d2 < dbest) || (d2 == dbest && m2 < mbest);
    dbest = take ? d2 : dbest;
    mbest = take ? m2 : mbest;
    if (!hi) {  // after all WMMAs; divergence is safe here
      size_t o = (size_t)b * Nq + (size_t)(qbase + t * 16 + col);
      outd[o] = dbest + sqq[t];
      outi[o] = mbest;
    }
  }
}

extern "C" void kernel_launch(void* const* d_in, const int* in_sizes, int n_in,
                              void* d_out, int out_size, void* d_ws,
                              size_t ws_size, hipStream_t stream) {
  const float* xyz1 = (const float*)d_in[0];  // [B,N,3]
  const float* xyz2 = (const float*)d_in[1];  // [B,M,3]
  const int B = 8;
  const int N = in_sizes[0] / (3 * B);
  const int M = in_sizes[1] / (3 * B);

  // Workspace: prepped A-operands (4 floats / point).
  float* A2 = (float*)d_ws;                 // B*M*4 floats (xyz2 as database)
  float* A1 = A2 + (size_t)B * M * 4;       // B*N*4 floats (xyz1 as database)

  // Outputs concatenated: dist1[B,N], dist2[B,M], idx1[B,N], idx2[B,M]
  float* out = (float*)d_out;
  float* dist1 = out;
  float* dist2 = out + (size_t)B * N;
  int* idx1 = (int*)(out + (size_t)B * N + (size_t)B * M);
  int* idx2 = idx1 + (size_t)B * N;

  int tot2 = B * M, tot1 = B * N;
  chamfer_prep<<<(tot2 + 255) / 256, 256, 0, stream>>>(xyz2, A2, tot2);
  chamfer_prep<<<(tot1 + 255) / 256, 256, 0, stream>>>(xyz1, A1, tot1);

  // Direction 1: queries = xyz1, database = xyz2 -> dist1/idx1
  int waves1 = B * (N / (16 * QT));
  chamfer_nn<<<waves1 / 8, 256, 0, stream>>>(A2, xyz1, dist1, idx1, M, N);
  // Direction 2: queries = xyz2, database = xyz1 -> dist2/idx2
  int waves2 = B * (M / (16 * QT));
  chamfer_nn<<<waves2 / 8, 256, 0, stream>>>(A1, xyz2, dist2, idx2, N, M);
}